// WindowedSelfAttention_34703335751905
// MI455X (gfx1250) — compile-verified
//
#include <hip/hip_runtime.h>

// Windowed self-attention, offsets -5..+5 (circular), B=4 H=16 S=4096 D=128, fp32.
// One wave32 per 16-query tile. Scores via V_WMMA_F32_16X16X4_F32 (two 16x16
// tiles cover the 11-tap band), softmax + weighted V-sum via LDS.

typedef __attribute__((ext_vector_type(2))) float v2f;
typedef __attribute__((ext_vector_type(4))) float v4f;
typedef __attribute__((ext_vector_type(8))) float v8f;

#define S_LEN 4096
#define SMASK 4095
#define DDIM 128
#define NTAP 11
#define SCALE 0.08838834764831845f  /* 1/sqrt(128) */

__launch_bounds__(32)
__global__ void wsa_wmma_kernel(const float* __restrict__ q,
                                const float* __restrict__ k,
                                const float* __restrict__ v,
                                float* __restrict__ out)
{
    __shared__ float smat[16 * 33];  // 16x32 score strip, stride 33 (bank-conflict pad)
    __shared__ float wmat[16 * 32];  // softmax weights, band entries only are read

    const int lane = threadIdx.x;     // 0..31, full wave active (WMMA needs EXEC all 1s)
    const int n    = lane & 15;
    const int hi   = lane >> 4;

    const int tile = blockIdx.x;             // over (B*H) * (S/16)
    const int bh   = tile >> 8;              // S/16 == 256
    const int s0   = (tile & 255) << 4;

    const float* qb = q + (size_t)bh * S_LEN * DDIM;
    const float* kb = k + (size_t)bh * S_LEN * DDIM;
    const float* vb = v + (size_t)bh * S_LEN * DDIM;
    float*       ob = out + (size_t)bh * S_LEN * DDIM;

    // Fragment row assignments (ISA layouts for 32-bit 16x4 A / 4x16 B):
    //   A: lanes 0-15 hold M=0..15 with K=0,1; lanes 16-31 same M with K=2,3.
    //   B: lanes 0-15 hold N=0..15 with K=0,1; lanes 16-31 same N with K=2,3.
    const int arow  = s0 + n;                    // Q rows (never wrap: tile-aligned)
    const int b0row = (s0 - 5 + n) & SMASK;      // K rows for tile0: s0-5 .. s0+10
    const int b1row = (s0 + 11 + n) & SMASK;     // K rows for tile1: s0+11 .. s0+26

    const float* aptr  = qb + (size_t)arow  * DDIM + 2 * hi;
    const float* b0ptr = kb + (size_t)b0row * DDIM + 2 * hi;
    const float* b1ptr = kb + (size_t)b1row * DDIM + 2 * hi;

    v8f acc0 = {};
    v8f acc1 = {};
#pragma unroll
    for (int c = 0; c < DDIM / 4; ++c) {
        v2f a  = *(const v2f*)(aptr  + 4 * c);
        v2f b0 = *(const v2f*)(b0ptr + 4 * c);
        v2f b1 = *(const v2f*)(b1ptr + 4 * c);
        acc0 = __builtin_amdgcn_wmma_f32_16x16x4_f32(false, a, false, b0,
                                                     (short)0, acc0, false, false);
        acc1 = __builtin_amdgcn_wmma_f32_16x16x4_f32(false, a, false, b1,
                                                     (short)0, acc1, false, false);
    }

    // Spill score tiles to LDS. C/D layout: VGPR r = rows M=r (lanes 0-15) and
    // M=r+8 (lanes 16-31), N = lane&15.
#pragma unroll
    for (int r = 0; r < 8; ++r) {
        const int m = r + 8 * hi;
        smat[m * 33 + n]      = acc0[r];
        smat[m * 33 + 16 + n] = acc1[r];
    }
    __syncthreads();

    // Softmax over the 11-tap band. Query row m uses strip columns m..m+10.
    if (lane < 16) {
        const int m = lane;
        float sc[NTAP];
        float mx = -1e30f;
#pragma unroll
        for (int t = 0; t < NTAP; ++t) {
            sc[t] = smat[m * 33 + m + t];
            mx = fmaxf(mx, sc[t]);
        }
        float sum = 0.0f;
#pragma unroll
        for (int t = 0; t < NTAP; ++t) {
            sc[t] = __expf((sc[t] - mx) * SCALE);
            sum += sc[t];
        }
        const float inv = 1.0f / sum;
#pragma unroll
        for (int t = 0; t < NTAP; ++t)
            wmat[m * 32 + m + t] = sc[t] * inv;
    }
    __syncthreads();

    // Weighted V accumulation: lane owns columns 4*lane..4*lane+3 of all 16
    // output rows. Each of the 32 V rows in the band is loaded exactly once.
    v4f oacc[16] = {};
#pragma unroll
    for (int j = 0; j < 32; ++j) {
        const int vrow = (s0 - 5 + j) & SMASK;
        const v4f vv = *(const v4f*)(vb + (size_t)vrow * DDIM + 4 * lane);
#pragma unroll
        for (int m = 0; m < 16; ++m) {
            // strip column j is a valid tap for row m iff m <= j <= m+10
            if (j >= m && j <= m + 10) {
                const float w = wmat[m * 32 + j];
                oacc[m][0] += w * vv[0];
                oacc[m][1] += w * vv[1];
                oacc[m][2] += w * vv[2];
                oacc[m][3] += w * vv[3];
            }
        }
    }

#pragma unroll
    for (int m = 0; m < 16; ++m)
        *(v4f*)(ob + (size_t)(s0 + m) * DDIM + 4 * lane) = oacc[m];
}

extern "C" void kernel_launch(void* const* d_in, const int* in_sizes, int n_in,
                              void* d_out, int out_size, void* d_ws, size_t ws_size,
                              hipStream_t stream) {
    (void)in_sizes; (void)n_in; (void)out_size; (void)d_ws; (void)ws_size;
    const float* q = (const float*)d_in[0];
    const float* k = (const float*)d_in[1];
    const float* v = (const float*)d_in[2];
    float* out = (float*)d_out;

    // B*H = 64 head-slices, S/16 = 256 query tiles each -> 16384 wave32 blocks
    const int nblocks = 64 * 256;
    wsa_wmma_kernel<<<dim3(nblocks), dim3(32), 0, stream>>>(q, k, v, out);
}